// MemAttention_10496900071858
// MI455X (gfx1250) — compile-verified
//
#include <hip/hip_runtime.h>
#include <cmath>

// ---------------------------------------------------------------------------
// MemAttention for MI455X (gfx1250, wave32).
// Roofline: ~350MB mandatory traffic (key_buffer 201MB + value_buffer 100MB +
// fc/fm/q/out) -> ~15us floor at 23.3 TB/s; total matrix FLOPs only ~7.2GF.
// => bandwidth-bound. Projections use true-fp32 V_WMMA_F32_16X16X4_F32
// (weights are L2-resident, high reuse); attention phase is a fused one-pass
// stream with b128 (float4) loads/stores to maximize bytes per VMEM issue.
// ---------------------------------------------------------------------------

typedef __attribute__((ext_vector_type(2))) float v2f;
typedef __attribute__((ext_vector_type(4))) float v4f;
typedef __attribute__((ext_vector_type(8))) float v8f;

#define B_  8
#define L_  8
#define DK  256
#define DV  128
#define HS  48
#define WS  64
#define HW  (HS * WS)   // 3072

// out[b,e,n] = sum_d W[d,e] * in[b,d,n]   (n = h*WS + w, contiguous)
// One wave = 16(e) x 16(n) tile, fp32 WMMA, K-step 4. 4 waves/block -> 16x64.
// D is compile-time so the K-loop unrolls by 4 with no remainder path.
template <int D>
__global__ __launch_bounds__(128) void proj_gemm(const float* __restrict__ in,
                                                 const float* __restrict__ W,
                                                 float* __restrict__ out) {
  const int wave = threadIdx.x >> 5;
  const int lane = threadIdx.x & 31;
  const int half = lane >> 4;     // 0: lanes 0-15, 1: lanes 16-31
  const int r    = lane & 15;
  const int n0 = blockIdx.x * 64 + wave * 16;
  const int e0 = blockIdx.y * 16;
  const int b  = blockIdx.z;

  const float* __restrict__ inB = in + (size_t)b * D * HW;

  v8f c = {};
#pragma unroll 4
  for (int k0 = 0; k0 < D; k0 += 4) {
    const int ka = k0 + 2 * half;          // ISA 16x4 f32 A layout:
    v2f a, bb;                             //  v0 <- K={0,2}, v1 <- K={1,3}
    a.x  = W[(size_t)(ka    ) * D + e0 + r];   // A[m=e][k=d] = W[d][e]
    a.y  = W[(size_t)(ka + 1) * D + e0 + r];
    bb.x = inB[(size_t)(ka    ) * HW + n0 + r];  // B[k][n] = in[b][d][n]
    bb.y = inB[(size_t)(ka + 1) * HW + n0 + r];
    // 8 args: (neg_a, A, neg_b, B, c_mod, C, reuse_a, reuse_b)
    c = __builtin_amdgcn_wmma_f32_16x16x4_f32(false, a, false, bb,
                                              (short)0, c, false, false);
  }

  float* __restrict__ outB = out + (size_t)b * D * HW;
#pragma unroll
  for (int v = 0; v < 8; ++v) {
    // C/D layout: VGPR v -> M = v (lanes 0-15) / v+8 (lanes 16-31), N = r
    outB[(size_t)(e0 + v + 8 * half) * HW + n0 + r] = c[v];
  }
}

// One block per (b,h): 256 threads = (w4 in [0,16)) x (sub in [0,16)).
// Each thread owns 4 consecutive w positions -> all KV streaming is b128.
// subs split the Dk reduction (16 d's each) and the Dv combine loop (8 e's).
__global__ __launch_bounds__(256) void mem_attention(
    const float* __restrict__ q,    const float* __restrict__ newk,
    const float* __restrict__ newv, const float* __restrict__ fm,
    const float* __restrict__ kbuf, const float* __restrict__ vbuf,
    float* __restrict__ out, float scale) {
  const int tid = threadIdx.x;
  const int w4  = tid & 15;                 // w = 4*w4 .. 4*w4+3
  const int sub = tid >> 4;                 // 0..15
  const int b   = blockIdx.x / HS;
  const int h   = blockIdx.x % HS;
  const int wb  = w4 * 4;

  __shared__ float part[16][L_][WS];        // 32 KB
  __shared__ float attnS[L_][WS];           // 2 KB

  // ---- scores: acc[l] = sum_d q[b,d,h,w] * kb[b,l,d,h,w] (sub splits d) ----
  v4f acc[L_];
#pragma unroll
  for (int l = 0; l < L_; ++l) acc[l] = (v4f){0.f, 0.f, 0.f, 0.f};

  const int d0 = sub * (DK / 16);
  const size_t rowOff = (size_t)h * WS + wb;
  for (int d = d0; d < d0 + DK / 16; ++d) {
    const v4f qv = *(const v4f*)(q + ((size_t)b * DK + d) * HW + rowOff);
#pragma unroll
    for (int l = 0; l < L_; ++l) {
      v4f kv;
      if (l < L_ - 1)   // FIFO shift: kb[:,l] = key_buffer[:,l+1]
        kv = *(const v4f*)(kbuf + (((size_t)b * L_ + (l + 1)) * DK + d) * HW + rowOff);
      else              // kb[:,L-1] = new_k
        kv = *(const v4f*)(newk + ((size_t)b * DK + d) * HW + rowOff);
      acc[l] += qv * kv;
    }
  }
#pragma unroll
  for (int l = 0; l < L_; ++l) *(v4f*)&part[sub][l][wb] = acc[l];
  __syncthreads();

  // reduce 16 partials -> raw scores (512 items, 2 per thread)
#pragma unroll
  for (int idx = tid; idx < L_ * WS; idx += 256) {
    const int l = idx >> 6, w = idx & 63;
    float s = 0.f;
#pragma unroll
    for (int p = 0; p < 16; ++p) s += part[p][l][w];
    attnS[l][w] = scale * s;
  }
  __syncthreads();

  // ---- softmax over WIDTH axis (64 values per (b,l,h)); wave tid>>5 owns l --
  {
    const int l    = tid >> 5;              // 8 waves, one l each (wave32)
    const int lane = tid & 31;
    float v0 = attnS[l][lane], v1 = attnS[l][lane + 32];
    float m = fmaxf(v0, v1);
#pragma unroll
    for (int off = 16; off > 0; off >>= 1)
      m = fmaxf(m, __shfl_xor(m, off, 32));
    float e0 = __expf(v0 - m), e1 = __expf(v1 - m);
    float s = e0 + e1;
#pragma unroll
    for (int off = 16; off > 0; off >>= 1)
      s += __shfl_xor(s, off, 32);
    const float inv = 1.f / s;
    attnS[l][lane]      = e0 * inv;
    attnS[l][lane + 32] = e1 * inv;
  }
  __syncthreads();

  v4f aw[L_];
#pragma unroll
  for (int l = 0; l < L_; ++l) aw[l] = *(const v4f*)&attnS[l][wb];

  // ---- combine: out[b,e,h,w] = fm + sum_l attn[l]*vb[b,l,e,h,w] (sub splits e)
#pragma unroll
  for (int e = sub * (DV / 16); e < (sub + 1) * (DV / 16); ++e) {
    v4f a = (v4f){0.f, 0.f, 0.f, 0.f};
#pragma unroll
    for (int l = 0; l < L_; ++l) {
      v4f vv;
      if (l < L_ - 1)
        vv = *(const v4f*)(vbuf + (((size_t)b * L_ + (l + 1)) * DV + e) * HW + rowOff);
      else
        vv = *(const v4f*)(newv + ((size_t)b * DV + e) * HW + rowOff);
      a += aw[l] * vv;
    }
    const size_t oidx = ((size_t)b * DV + e) * HW + rowOff;
    const v4f f = *(const v4f*)(fm + oidx);
    *(v4f*)(out + oidx) = f + a;            // ALPHA = 1.0
  }
}

extern "C" void kernel_launch(void* const* d_in, const int* in_sizes, int n_in,
                              void* d_out, int out_size, void* d_ws, size_t ws_size,
                              hipStream_t stream) {
  (void)in_sizes; (void)n_in; (void)out_size; (void)ws_size;
  const float* fc   = (const float*)d_in[0];
  const float* fm   = (const float*)d_in[1];
  const float* kbuf = (const float*)d_in[2];
  const float* vbuf = (const float*)d_in[3];
  const float* Q    = (const float*)d_in[4];
  const float* K    = (const float*)d_in[5];
  const float* V    = (const float*)d_in[6];
  float* out = (float*)d_out;

  // workspace: q (25.2MB) | new_k (25.2MB) | new_v (12.6MB)
  float* q    = (float*)d_ws;
  float* newk = q    + (size_t)B_ * DK * HW;
  float* newv = newk + (size_t)B_ * DK * HW;

  proj_gemm<DK><<<dim3(HW / 64, DK / 16, B_), dim3(128), 0, stream>>>(fc, Q, q);
  proj_gemm<DK><<<dim3(HW / 64, DK / 16, B_), dim3(128), 0, stream>>>(fc, K, newk);
  proj_gemm<DV><<<dim3(HW / 64, DV / 16, B_), dim3(128), 0, stream>>>(fm, V, newv);

  // SCALE = log(L*Hs*Ws + Hs*Ws) / log(1000) / sqrt(Dk)
  const float scale = (float)(std::log(27648.0) / std::log(1000.0) / 16.0);
  mem_attention<<<dim3(B_ * HS), dim3(256), 0, stream>>>(
      q, newk, newv, fm, kbuf, vbuf, out, scale);
}